// dblp_hnode_prompt_layer_feature_weighted_sum_21534966022312
// MI455X (gfx1250) — compile-verified
//
#include <hip/hip_runtime.h>
#include <hip/hip_bf16.h>
#include <stdint.h>

// GNN scatter-add workload: elu elementwise + two segment_sums over 800K edges.
// L2-resident (all arrays << 192MB); bound by L2 atomic/gather throughput.
// Strategy: precompute emb once; wave32-native edge scatter (16 lanes/edge,
// float4 per lane, hw f32 atomics); async-to-LDS staging of index streams.

#define DFEAT 64
#define DFEAT4 16
#define WAVES_PER_BLOCK 8
#define TILE_EDGES 32  // edges staged per wave per pass

typedef __attribute__((address_space(1))) int gint;   // global
typedef __attribute__((address_space(3))) int lint;   // LDS

#if __has_builtin(__builtin_amdgcn_global_load_async_to_lds_b32)
#define HAVE_ASYNC_LDS 1
#else
#define HAVE_ASYNC_LDS 0
#endif

__device__ __forceinline__ void wait_async_zero() {
#if __has_builtin(__builtin_amdgcn_s_wait_asynccnt)
    __builtin_amdgcn_s_wait_asynccnt(0);
#else
    asm volatile("s_wait_asynccnt 0" ::: "memory");
#endif
}

__device__ __forceinline__ void wait_ds_zero() {
#if __has_builtin(__builtin_amdgcn_s_wait_dscnt)
    __builtin_amdgcn_s_wait_dscnt(0);
#else
    asm volatile("s_wait_dscnt 0" ::: "memory");
#endif
}

__device__ __forceinline__ void atomic_add_f32(float* p, float v) {
    // Should lower to GLOBAL_ATOMIC_ADD_F32 without return (STOREcnt).
    __hip_atomic_fetch_add(p, v, __ATOMIC_RELAXED, __HIP_MEMORY_SCOPE_AGENT);
}

__device__ __forceinline__ float elu1(float a) {
    return a > 0.0f ? a : (__expf(a) - 1.0f);
}

// Shared body for both scatter passes.
// ONEHOP=false: dmat[dst] += smat[src] * (1 + [e==0] + [e==4] + [e==5])
// ONEHOP=true : dmat[dst] += smat[src] * [e==3]
// Each wave stages a 32-edge index tile (src/dst/e) into its own LDS region
// via async-to-LDS (no barrier needed: wave consumes only its own tile), then
// processes 2 edges per pass: 16 lanes per edge, one float4 per lane
// (global_load_b128 gather + 4 hw f32 atomics).
template <bool ONEHOP>
__device__ __forceinline__ void edge_scatter_body(const int* __restrict__ src,
                                                  const int* __restrict__ dst,
                                                  const int* __restrict__ ef,
                                                  const float* __restrict__ smat,
                                                  float* __restrict__ dmat,
                                                  int n_edges) {
    __shared__ int stage[WAVES_PER_BLOCK * 3 * TILE_EDGES];
    const int wave = threadIdx.x >> 5;
    const int lane = threadIdx.x & 31;
    int* my = &stage[wave * 3 * TILE_EDGES];

    const int sub = lane >> 4;    // which of the 2 edges this half-wave owns
    const int d4i = lane & 15;    // float4 index within the 64-float row

    const int ntiles = (n_edges + TILE_EDGES - 1) / TILE_EDGES;
    for (int tile = blockIdx.x * WAVES_PER_BLOCK + wave; tile < ntiles;
         tile += gridDim.x * WAVES_PER_BLOCK) {
        const int e0 = tile * TILE_EDGES;
        const int nrem = n_edges - e0;                 // >= 1
        const int cl = lane < nrem ? lane : nrem - 1;  // clamp tail
        const int eid = e0 + cl;

        // WAR: prior passes' ds_loads must drain before async writes land.
        wait_ds_zero();
#if HAVE_ASYNC_LDS
        __builtin_amdgcn_global_load_async_to_lds_b32(
            (gint*)&src[eid], (lint*)&my[lane], 0, 0);
        __builtin_amdgcn_global_load_async_to_lds_b32(
            (gint*)&dst[eid], (lint*)&my[TILE_EDGES + lane], 0, 0);
        __builtin_amdgcn_global_load_async_to_lds_b32(
            (gint*)&ef[eid], (lint*)&my[2 * TILE_EDGES + lane], 0, 0);
        wait_async_zero();
#else
        my[lane] = src[eid];
        my[TILE_EDGES + lane] = dst[eid];
        my[2 * TILE_EDGES + lane] = ef[eid];
#endif

#pragma unroll 4
        for (int p = 0; p < TILE_EDGES / 2; ++p) {
            const int le = p * 2 + sub;  // local edge 0..31
            if (le >= nrem) continue;
            const int s  = my[le];
            const int dd = my[TILE_EDGES + le];
            const int ee = my[2 * TILE_EDGES + le];

            float scale;
            if (ONEHOP) {
                if (ee != 3) continue;  // ~5/6 of edges masked out cheaply
                scale = 1.0f;
            } else {
                scale = 1.0f + (ee == 0 ? 1.0f : 0.0f)
                             + (ee == 4 ? 1.0f : 0.0f)
                             + (ee == 5 ? 1.0f : 0.0f);
            }

            const float4 v =
                ((const float4*)(smat + (size_t)s * DFEAT))[d4i];  // b128 gather
            float* dp = dmat + (size_t)dd * DFEAT + d4i * 4;
            atomic_add_f32(dp + 0, v.x * scale);
            atomic_add_f32(dp + 1, v.y * scale);
            atomic_add_f32(dp + 2, v.z * scale);
            atomic_add_f32(dp + 3, v.w * scale);
        }
    }
}

// Defined FIRST so the disasm snippet shows this (perf-critical) kernel.
__global__ void edge_scatter_twohop(const int* __restrict__ src,
                                    const int* __restrict__ dst,
                                    const int* __restrict__ ef,
                                    const float* __restrict__ smat,
                                    float* __restrict__ dmat,
                                    int n_edges) {
    edge_scatter_body<false>(src, dst, ef, smat, dmat, n_edges);
}

__global__ void edge_scatter_onehop(const int* __restrict__ src,
                                    const int* __restrict__ dst,
                                    const int* __restrict__ ef,
                                    const float* __restrict__ smat,
                                    float* __restrict__ dmat,
                                    int n_edges) {
    edge_scatter_body<true>(src, dst, ef, smat, dmat, n_edges);
}

// Fused: emb = elu(x * broadcast(w)); ft2 = 0; out = 0.  Vectorized float4.
__global__ void init_elu_zero(const float* __restrict__ x,
                              const float* __restrict__ w,
                              float* __restrict__ emb,
                              float* __restrict__ ft2,
                              float* __restrict__ out,
                              int n4) {
    const float4* x4 = (const float4*)x;
    const float4* w4 = (const float4*)w;
    float4* e4 = (float4*)emb;
    float4* f4 = (float4*)ft2;
    float4* o4 = (float4*)out;
    float4 z; z.x = z.y = z.z = z.w = 0.0f;
    for (int i = blockIdx.x * blockDim.x + threadIdx.x; i < n4;
         i += gridDim.x * blockDim.x) {
        float4 xv = x4[i];
        float4 wv = w4[i & (DFEAT4 - 1)];  // D/4 == 16 float4s per row
        float4 r;
        r.x = elu1(xv.x * wv.x);
        r.y = elu1(xv.y * wv.y);
        r.z = elu1(xv.z * wv.z);
        r.w = elu1(xv.w * wv.w);
        e4[i] = r;
        f4[i] = z;
        o4[i] = z;
    }
}

extern "C" void kernel_launch(void* const* d_in, const int* in_sizes, int n_in,
                              void* d_out, int out_size, void* d_ws, size_t ws_size,
                              hipStream_t stream) {
    const float* x  = (const float*)d_in[0];  // graph_embedding [N,64]
    const float* w  = (const float*)d_in[1];  // weight [1,64]
    const int* src  = (const int*)d_in[2];    // [E]
    const int* dst  = (const int*)d_in[3];    // [E]
    const int* ef   = (const int*)d_in[4];    // [E]
    float* out = (float*)d_out;               // [N,64]

    const int nd = in_sizes[0];   // N*64
    const int E  = in_sizes[2];

    float* emb = (float*)d_ws;        // 12.8 MB
    float* ft2 = emb + (size_t)nd;    // 12.8 MB (two-hop accumulator)

    const int n4 = nd / 4;
    int ib = (n4 + 255) / 256;
    if (ib > 2048) ib = 2048;
    init_elu_zero<<<ib, 256, 0, stream>>>(x, w, emb, ft2, out, n4);

    const int ntiles = (E + TILE_EDGES - 1) / TILE_EDGES;
    int sb = (ntiles + WAVES_PER_BLOCK - 1) / WAVES_PER_BLOCK;  // 3125 blocks
    // Two-hop: ft2[dst] += emb[src] * scale(e)
    edge_scatter_twohop<<<sb, 256, 0, stream>>>(src, dst, ef, emb, ft2, E);
    // One-hop: out[dst] += ft2[src] * [e==3]   (needs ft2 complete -> new launch)
    edge_scatter_onehop<<<sb, 256, 0, stream>>>(src, dst, ef, ft2, out, E);
}